// CSAM_5145370821395
// MI455X (gfx1250) — compile-verified
//
#include <hip/hip_runtime.h>
#include <hip/hip_bf16.h>

typedef float v2f __attribute__((ext_vector_type(2)));
typedef float v8f __attribute__((ext_vector_type(8)));

#define BB 32
#define CC 64
#define NN 36864            // 192*192
#define KSPLIT 36           // K-chunks per batch for the Gram kernel
#define K3_BLOCKS 36        // blocks per batch for the output kernel

__device__ __forceinline__ v8f wmma4(v2f a, v2f b, v8f c) {
    // D(16x16,f32) = A(16x4,f32) * B(4x16,f32) + C
    return __builtin_amdgcn_wmma_f32_16x16x4_f32(
        /*neg_a=*/false, a, /*neg_b=*/false, b,
        /*c_mod=*/(short)0, c, /*reuse_a=*/false, /*reuse_b=*/false);
}

// ---------------------------------------------------------------- zero init
__global__ void CSAM_zero_kernel(float* __restrict__ p, int n) {
    int i = blockIdx.x * blockDim.x + threadIdx.x;
    if (i < n) p[i] = 0.0f;
}

// ---------------------------------------------------------------- Gram: energy[b] = q_b q_b^T  (symmetric: upper triangle only)
__global__ __launch_bounds__(256) void CSAM_gram_kernel(const float* __restrict__ x,
                                                        float* __restrict__ energy) {
    const int b    = blockIdx.x;        // batch
    const int ks   = blockIdx.y;        // K-chunk
    const int lane = threadIdx.x & 31;
    const int wave = threadIdx.x >> 5;  // 0..7
    const int lhalf = lane >> 4;        // 0/1
    const int lmod  = lane & 15;

    __shared__ float lds[CC * CC];      // 16 KB partial Gram
    for (int i = threadIdx.x; i < CC * CC; i += 256) lds[i] = 0.0f;
    __syncthreads();

    const int kPerBlock = NN / KSPLIT;      // 1024
    const int kPerWave  = kPerBlock / 8;    // 128
    const int k0 = ks * kPerBlock + wave * kPerWave;

    const float* qb = x + (size_t)b * CC * NN;

    // only gi <= gj (upper triangle) is used; the rest is dead-stripped
    v8f acc[4][4];
    const v8f vzero = {};
#pragma unroll
    for (int i = 0; i < 4; ++i)
#pragma unroll
        for (int j = 0; j < 4; ++j) acc[i][j] = vzero;

    // each k-step covers K=4; A and B fragments of q share the same lane layout
    for (int kk = 0; kk < kPerWave; kk += 4) {
        const int kbase = k0 + kk + 2 * lhalf;
        v2f frag[4];
#pragma unroll
        for (int g = 0; g < 4; ++g) {
            const float* p = qb + (size_t)(16 * g + lmod) * NN + kbase;
            frag[g] = *(const v2f*)p;   // 8B aligned
        }
#pragma unroll
        for (int gi = 0; gi < 4; ++gi)
#pragma unroll
            for (int gj = gi; gj < 4; ++gj)
                acc[gi][gj] = wmma4(frag[gi], frag[gj], acc[gi][gj]);
    }

    // reduce the 8 per-wave partial Grams into LDS (ds_add_f32);
    // off-diagonal tiles scatter to both (row,col) and the mirrored (col,row)
#pragma unroll
    for (int gi = 0; gi < 4; ++gi)
#pragma unroll
        for (int gj = gi; gj < 4; ++gj)
#pragma unroll
            for (int r = 0; r < 8; ++r) {
                const int m = r + 8 * lhalf;           // row within tile
                const int n = lmod;                    // col within tile
                const float v = acc[gi][gj][r];
                __hip_atomic_fetch_add(&lds[(16 * gi + m) * CC + (16 * gj + n)], v,
                                       __ATOMIC_RELAXED, __HIP_MEMORY_SCOPE_WORKGROUP);
                if (gi != gj)
                    __hip_atomic_fetch_add(&lds[(16 * gj + n) * CC + (16 * gi + m)], v,
                                           __ATOMIC_RELAXED, __HIP_MEMORY_SCOPE_WORKGROUP);
            }
    __syncthreads();

    // one global f32 atomic-add per element per workgroup
    float* eb = energy + (size_t)b * CC * CC;
    for (int i = threadIdx.x; i < CC * CC; i += 256)
        __hip_atomic_fetch_add(&eb[i], lds[i],
                               __ATOMIC_RELAXED, __HIP_MEMORY_SCOPE_AGENT);
}

// ---------------------------------------------------------------- softmax(rowmax - e) == exp(min-e)/sum
__global__ void CSAM_softmax_kernel(float* __restrict__ energy) {
    const int row = blockIdx.x * blockDim.x + threadIdx.x;   // 0 .. B*C-1
    if (row >= BB * CC) return;
    float* e = energy + (size_t)row * CC;
    float v[CC];
#pragma unroll
    for (int i = 0; i < CC; ++i) v[i] = e[i];
    float mn = v[0];
#pragma unroll
    for (int i = 1; i < CC; ++i) mn = fminf(mn, v[i]);
    float s = 0.0f;
#pragma unroll
    for (int i = 0; i < CC; ++i) { v[i] = __expf(mn - v[i]); s += v[i]; }
    const float inv = __frcp_rn(s);
#pragma unroll
    for (int i = 0; i < CC; ++i) e[i] = v[i] * inv;
}

// ---------------------------------------------------------------- out = att * q, fused y = x*(1 + g*out)
__global__ __launch_bounds__(256) void CSAM_out_kernel(const float* __restrict__ x,
                                                       const float* __restrict__ att,
                                                       const float* __restrict__ gamma,
                                                       float* __restrict__ y) {
    const int b    = blockIdx.y;
    const int lane = threadIdx.x & 31;
    const int wave = threadIdx.x >> 5;
    const int lhalf = lane >> 4;
    const int lmod  = lane & 15;
    const float g = gamma[0];

    // resident A fragments: entire 64x64 attention matrix for this batch
    const float* ab = att + (size_t)b * CC * CC;
    v2f afrag[4][16];
#pragma unroll
    for (int cg = 0; cg < 4; ++cg)
#pragma unroll
        for (int s = 0; s < 16; ++s) {
            const float* p = ab + (size_t)(16 * cg + lmod) * CC + 4 * s + 2 * lhalf;
            afrag[cg][s] = *(const v2f*)p;
        }

    const float* qb = x + (size_t)b * CC * NN;
    float*       yb = y + (size_t)b * CC * NN;

    const int nTiles = NN / 16;                     // 2304
    const int wavesPerBatch = K3_BLOCKS * 8;        // 288 -> exactly 8 tiles per wave
    const int waveGlobal = blockIdx.x * 8 + wave;

    for (int t = waveGlobal; t < nTiles; t += wavesPerBatch) {
        const int n0 = t * 16;
        v8f acc[4];
        const v8f vzero = {};
#pragma unroll
        for (int cg = 0; cg < 4; ++cg) acc[cg] = vzero;

#pragma unroll
        for (int s = 0; s < 16; ++s) {
            const int d0 = 4 * s + 2 * lhalf;
            const float* p = qb + (size_t)d0 * NN + n0 + lmod;
            v2f bfrag;
            bfrag.x = p[0];
            bfrag.y = p[NN];
#pragma unroll
            for (int cg = 0; cg < 4; ++cg)
                acc[cg] = wmma4(afrag[cg][s], bfrag, acc[cg]);
        }

        // epilogue: y = x * (g*out) + x
#pragma unroll
        for (int cg = 0; cg < 4; ++cg)
#pragma unroll
            for (int r = 0; r < 8; ++r) {
                const int c = 16 * cg + r + 8 * lhalf;
                const size_t idx = (size_t)c * NN + n0 + lmod;
                const float xv = qb[idx];
                yb[idx] = fmaf(xv, g * acc[cg][r], xv);
            }
    }
}

extern "C" void kernel_launch(void* const* d_in, const int* in_sizes, int n_in,
                              void* d_out, int out_size, void* d_ws, size_t ws_size,
                              hipStream_t stream) {
    (void)in_sizes; (void)n_in; (void)out_size; (void)ws_size;
    const float* x     = (const float*)d_in[0];
    const float* gamma = (const float*)d_in[1];
    float* y      = (float*)d_out;
    float* energy = (float*)d_ws;                 // 32*64*64 floats = 512 KB

    const int esz = BB * CC * CC;
    CSAM_zero_kernel<<<(esz + 255) / 256, 256, 0, stream>>>(energy, esz);
    CSAM_gram_kernel<<<dim3(BB, KSPLIT), 256, 0, stream>>>(x, energy);
    CSAM_softmax_kernel<<<(BB * CC + 255) / 256, 256, 0, stream>>>(energy);
    CSAM_out_kernel<<<dim3(K3_BLOCKS, BB), 256, 0, stream>>>(x, energy, gamma, y);
}